// TimeVaryingDelayLine_33406255628322
// MI455X (gfx1250) — compile-verified
//
#include <hip/hip_runtime.h>
#include <stdint.h>

// Problem constants (from the reference setup)
#define MAXD     40000
#define T_LEN    1000000
#define BATCHES  8
#define CHANS    2
#define N_SERIES (BATCHES * CHANS)

// Tiling: one block stages x_pad[t0 .. t0+MAXD+TILE) into LDS.
// WIN = 80000 floats = 320,000 bytes <= 320KB CDNA5 workgroup LDS cap.
#define TILE   40000
#define WIN    (MAXD + TILE)
#define BLOCK  1024

// gfx1250 async global->LDS path (guarded so either toolchain compiles)
#if defined(__has_builtin)
#  if __has_builtin(__builtin_amdgcn_global_load_async_to_lds_b128)
#    define HAVE_ASYNC_LDS 1
#  endif
#endif
#ifndef HAVE_ASYNC_LDS
#  define HAVE_ASYNC_LDS 0
#endif

// Builtin parameter types per hipcc diagnostic:
//   arg0: __attribute__((vector_size(16))) int  __device__ * (global, AS1)
//   arg1: LDS-side pointer (AS3), same vector pointee
typedef int v4i __attribute__((vector_size(4 * sizeof(int))));
typedef __attribute__((address_space(1))) v4i gv4i;   // global int4*
typedef __attribute__((address_space(3))) v4i lv4i;   // LDS int4*

__global__ __launch_bounds__(BLOCK, 1)
void TimeVaryingDelayLine_33406255628322_kernel(const float* __restrict__ x,
                                                const float* __restrict__ dt,
                                                const float* __restrict__ buf,
                                                float* __restrict__ y)
{
    extern __shared__ float smem[];  // WIN floats (dynamic LDS, 320,000 B)

    const int series = blockIdx.y;
    const int t0     = blockIdx.x * TILE;
    const int tile   = (T_LEN - t0 < TILE) ? (T_LEN - t0) : TILE;

    const float* __restrict__ xs = x   + (size_t)series * T_LEN;
    const float* __restrict__ ds = dt  + (size_t)series * T_LEN;
    const float* __restrict__ bs = buf + (size_t)series * MAXD;
    float*       __restrict__ ys = y   + (size_t)series * T_LEN;

    // ---- Stage x_pad[t0 .. t0+WIN) into LDS; x_pad = [buffer | x] --------
    // 16B chunks: t0 % 4 == 0 and MAXD % 4 == 0, so no chunk straddles the
    // buffer/x boundary and both global and LDS addresses are 16B aligned.
    for (int c = threadIdx.x; c < WIN / 4; c += BLOCK) {
        const int g = t0 + c * 4;                 // x_pad coordinate
        if (g >= MAXD + T_LEN) break;             // (never hits: exact tiling)
        const float* src = (g < MAXD) ? (bs + g) : (xs + (g - MAXD));
#if HAVE_ASYNC_LDS
        __builtin_amdgcn_global_load_async_to_lds_b128(
            (gv4i*)src, (lv4i*)(smem + c * 4), 0, 0);
#else
        float4 v = *(const float4*)src;
        *(float4*)(smem + c * 4) = v;
#endif
    }
#if HAVE_ASYNC_LDS
#  if defined(__has_builtin) && __has_builtin(__builtin_amdgcn_s_wait_asynccnt)
    __builtin_amdgcn_s_wait_asynccnt(0);
#  else
    asm volatile("s_wait_asynccnt 0" ::: "memory");
#  endif
#endif
    __syncthreads();

    // ---- Fractional-delay gather + lerp from LDS -------------------------
    // idx0 = MAXD + t - floor(dt); local offset o0 = idx0 - t0.
    for (int j = (int)threadIdx.x * 4; j < tile; j += BLOCK * 4) {
        float4 d4 = *(const float4*)(ds + t0 + j);
        float dv[4] = {d4.x, d4.y, d4.z, d4.w};
        float r[4];
#pragma unroll
        for (int k = 0; k < 4; ++k) {
            float dvk = dv[k];
            int   i0  = (int)dvk;                 // floor (dvk >= 0)
            float fr  = dvk - (float)i0;
            int   o0  = (j + k) + MAXD - i0;      // tap at delay floor(dt)
            int   o1  = o0 - 1;                   // tap at delay floor(dt)+1
            o1 = (o1 < 0) ? 0 : o1;               // matches reference clamp
            float tap0 = smem[o0];
            float tap1 = smem[o1];
            r[k] = (1.0f - fr) * tap0 + fr * tap1;
        }
        float4 out = make_float4(r[0], r[1], r[2], r[3]);
        *(float4*)(ys + t0 + j) = out;
    }
}

extern "C" void kernel_launch(void* const* d_in, const int* in_sizes, int n_in,
                              void* d_out, int out_size, void* d_ws, size_t ws_size,
                              hipStream_t stream) {
    (void)in_sizes; (void)n_in; (void)out_size; (void)d_ws; (void)ws_size;

    const float* x   = (const float*)d_in[0];   // (8, 2, 1e6) f32
    const float* dt  = (const float*)d_in[1];   // (8, 2, 1e6) f32
    const float* buf = (const float*)d_in[2];   // (8, 2, 40000) f32
    float*       y   = (float*)d_out;           // (8, 2, 1e6) f32

    const size_t smem_bytes = (size_t)WIN * sizeof(float);  // 320,000 B

    // Opt in to >default dynamic LDS (idempotent, deterministic, non-stream op)
    (void)hipFuncSetAttribute(
        reinterpret_cast<const void*>(TimeVaryingDelayLine_33406255628322_kernel),
        hipFuncAttributeMaxDynamicSharedMemorySize, (int)smem_bytes);

    dim3 grid((T_LEN + TILE - 1) / TILE, N_SERIES);   // (25, 16)
    TimeVaryingDelayLine_33406255628322_kernel<<<grid, BLOCK, smem_bytes, stream>>>(
        x, dt, buf, y);
}